// TplinkerHandshakingKernelPlus_72206990180359
// MI455X (gfx1250) — compile-verified
//
#include <hip/hip_runtime.h>

typedef __attribute__((ext_vector_type(2))) float v2f;
typedef __attribute__((ext_vector_type(4))) float v4f;
typedef __attribute__((ext_vector_type(8))) float v8f;

#define B_    8
#define S_    384
#define H_    256
#define N2_   512                      // 2*H (combined pg|pv projection width)
#define M_    (B_ * S_)                // 3072 rows of X
#define NPAIR ((S_ * (S_ + 1)) / 2)    // 73920 upper-triangular pairs

// ---------------------------------------------------------------------------
// Stage 1: P[m, n] = sum_k X[m,k] * W[n & 255, (n>=256 ? 256 : 0) + k]
// (n < 256 -> pg = X·Wg^T, n >= 256 -> pv = X·Wv^T, fused into one GEMM)
// One wave32 per 16x16 output tile, V_WMMA_F32_16X16X4_F32, K-loop of 64.
//
// A f32 16x4 layout (ISA 7.12.2): lanes 0-15 hold M=0..15 with K=0 (v0), K=1
// (v1); lanes 16-31 hold K=2,3  -> per-lane contiguous float2 of X.
// B f32 4x16 mirrored: lane = N, lanes 0-15 hold K=0,1; lanes 16-31 K=2,3
// -> per-lane contiguous float2 of W's row (contraction is W's column dim).
// C/D: VGPR r -> M = r + 8*(lane>=16), N = lane&15.
// ---------------------------------------------------------------------------
__global__ __launch_bounds__(128) void proj_wmma_f32(const float* __restrict__ X,
                                                     const float* __restrict__ W,
                                                     float* __restrict__ P) {
  const int wave  = threadIdx.x >> 5;
  const int lane  = threadIdx.x & 31;
  const int lrow  = lane & 15;
  const int lhalf = lane >> 4;

  const int NT   = N2_ / 16;                 // 32 N-tiles
  const int tile = blockIdx.x * 4 + wave;    // 192*32 = 6144 tiles total
  const int mt   = tile / NT;
  const int nt   = tile - mt * NT;

  const int m = mt * 16 + lrow;
  const int n = nt * 16 + lrow;

  // Per-lane base pointers; k advances by 4 per WMMA, this lane covers
  // K = k0 + 2*lhalf + {0,1}.
  const float* __restrict__ arow = X + (size_t)m * H_ + lhalf * 2;
  const float* __restrict__ brow = W + (size_t)(n & (H_ - 1)) * N2_ +
                                   ((n >= H_) ? H_ : 0) + lhalf * 2;

  v8f acc = {};
#pragma unroll 8
  for (int k0 = 0; k0 < H_; k0 += 4) {
    v2f a = *(const v2f*)(arow + k0);
    v2f b = *(const v2f*)(brow + k0);
    // 8 args: (neg_a, A, neg_b, B, c_mod, C, reuse_a, reuse_b)
    acc = __builtin_amdgcn_wmma_f32_16x16x4_f32(false, a, false, b,
                                                (short)0, acc, false, false);
  }

  // Scatter D per documented C/D layout.
  float* __restrict__ prow = P + (size_t)(mt * 16 + lhalf * 8) * N2_ +
                             nt * 16 + lrow;
#pragma unroll
  for (int r = 0; r < 8; ++r) {
    prow[(size_t)r * N2_] = acc[r];
  }
}

// ---------------------------------------------------------------------------
// Stage 2: out[b, p, o] = relu(P[b*S+i, o] + P[b*S+j, 256+o] + bias[o])
// where p -> (i, j) enumerates the upper triangle row-major (i <= j).
// One block (256 threads) per pair; uniform decode; float4 non-temporal
// stores; 8 batches covered as 4 batch-groups x 2 iterations.
// Reads hit L2 (P is 6 MB); stores stream 605 MB to HBM -> bound at 23.3 TB/s.
// ---------------------------------------------------------------------------
__global__ __launch_bounds__(256) void pair_relu_stream(const float* __restrict__ P,
                                                        const float* __restrict__ bias,
                                                        float* __restrict__ out) {
  const int p = blockIdx.x;

  // start(i) = i*S - i*(i-1)/2 ; invert with sqrt then exact integer fix-up.
  int i = (int)(((float)(2 * S_ + 1) -
                 sqrtf((float)((2 * S_ + 1) * (2 * S_ + 1) - 8 * p))) * 0.5f);
  if (i < 0) i = 0;
  if (i > S_ - 1) i = S_ - 1;
  while (i > 0 && (i * S_ - (i * (i - 1)) / 2) > p) --i;
  while (((i + 1) * S_ - ((i + 1) * i) / 2) <= p) ++i;
  const int j = i + (p - (i * S_ - (i * (i - 1)) / 2));

  const int t    = threadIdx.x;
  const int o4   = (t & 63) << 2;   // 64 lanes * float4 = 256 columns
  const int bgrp = t >> 6;          // 4 batch groups

  const v4f bias4 = *(const v4f*)(bias + o4);

#pragma unroll
  for (int it = 0; it < 2; ++it) {
    const int b = it * 4 + bgrp;
    const v4f pg = *(const v4f*)(P + (size_t)(b * S_ + i) * N2_ + o4);
    const v4f pv = *(const v4f*)(P + (size_t)(b * S_ + j) * N2_ + H_ + o4);
    v4f r = pg + pv + bias4;
    r.x = fmaxf(r.x, 0.0f);
    r.y = fmaxf(r.y, 0.0f);
    r.z = fmaxf(r.z, 0.0f);
    r.w = fmaxf(r.w, 0.0f);
    const size_t oidx = ((size_t)b * NPAIR + (size_t)p) * H_ + o4;
    __builtin_nontemporal_store(r, (v4f*)(out + oidx));
  }
}

extern "C" void kernel_launch(void* const* d_in, const int* in_sizes, int n_in,
                              void* d_out, int out_size, void* d_ws, size_t ws_size,
                              hipStream_t stream) {
  const float* X    = (const float*)d_in[0];   // (8, 384, 256) f32
  const float* W    = (const float*)d_in[1];   // (256, 512) f32
  const float* bias = (const float*)d_in[2];   // (256,) f32
  float* out        = (float*)d_out;           // (8, 73920, 256) f32
  float* P          = (float*)d_ws;            // (3072, 512) f32 = 6 MB scratch

  // Stage 1: fused projection GEMM via WMMA. 6144 wave-tiles / 4 waves per blk.
  const int tiles = (M_ / 16) * (N2_ / 16);
  proj_wmma_f32<<<dim3(tiles / 4), dim3(128), 0, stream>>>(X, W, P);

  // Stage 2: triangular pair expansion, store-bandwidth bound.
  pair_relu_stream<<<dim3(NPAIR), dim3(256), 0, stream>>>(P, bias, out);
}